// SpringMass_41102837022981
// MI455X (gfx1250) — compile-verified
//
#include <hip/hip_runtime.h>
#include <math.h>

// ---- problem constants (match reference) ----
#define DT              5e-05f
#define NUM_SUBSTEPS    100
#define SPRING_Y        30000.0f
#define DASHPOT_DAMPING 100.0f
#define DRAG_DAMPING    1.0f
#define NV              100000
#define NS              1000000

struct __attribute__((aligned(8))) I2 { int a, b; };

// 64-byte vertex record: position(+inv_mass in w), velocity, force accum, pad.
// One random endpoint access (2 b128 loads + 3 f32 atomics) touches a single
// 128B cacheline instead of three separate arrays' lines.
struct __attribute__((aligned(16))) Vert {
    float4 x;    // xyz position, w = 1/mass (constant across substeps)
    float4 v;    // xyz velocity
    float4 f;    // xyz force accumulator (zeroed by integrate)
    float4 pad;  // pad to 64B so records never straddle cachelines
};

// ------------------------------------------------------------------
// Kernel 0: build padded AoS working state each launch (deterministic).
// ------------------------------------------------------------------
__global__ __launch_bounds__(256) void sm_init(const float* __restrict__ x0,
                                               const float* __restrict__ mass,
                                               Vert* __restrict__ W) {
    int i = blockIdx.x * blockDim.x + threadIdx.x;
    if (i >= NV) return;
    float inv_m = 1.0f / mass[i];
    float4 z4 = make_float4(0.f, 0.f, 0.f, 0.f);
    Vert w;
    w.x = make_float4(x0[3 * i + 0], x0[3 * i + 1], x0[3 * i + 2], inv_m);
    w.v = z4;
    w.f = z4;
    w.pad = z4;
    W[i] = w;
}

// ------------------------------------------------------------------
// Kernel 1: per-spring force pair. Each endpoint: two b128 loads and
// three global_atomic_add_f32, all within one 64B record (one line).
// ------------------------------------------------------------------
__global__ __launch_bounds__(256) void sm_springs(Vert* __restrict__ W,
                                                  const int* __restrict__ springs,
                                                  const float* __restrict__ rest) {
    int s = blockIdx.x * blockDim.x + threadIdx.x;
    if (s >= NS) return;

    I2 sp = ((const I2*)springs)[s];      // coalesced global_load_b64
    int i1 = sp.a, i2 = sp.b;
    float r = rest[s];                    // coalesced global_load_b32

    Vert* p1 = W + i1;
    Vert* p2 = W + i2;
    float4 x1 = p1->x;                    // b128 @ base+0
    float4 v1 = p1->v;                    // b128 @ base+16 (same line)
    float4 x2 = p2->x;
    float4 v2 = p2->v;

    float dx = x2.x - x1.x;
    float dy = x2.y - x1.y;
    float dz = x2.z - x1.z;
    float len = sqrtf(dx * dx + dy * dy + dz * dz);
    float inv_len = 1.0f / len;
    float dxn = dx * inv_len, dyn = dy * inv_len, dzn = dz * inv_len;

    float vrel = (v2.x - v1.x) * dxn + (v2.y - v1.y) * dyn + (v2.z - v1.z) * dzn;

    // scalar magnitude of (spring + dashpot) force along d
    float mag = SPRING_Y * (len / r - 1.0f) + DASHPOT_DAMPING * vrel;
    float fx = mag * dxn, fy = mag * dyn, fz = mag * dzn;

    // global_atomic_add_f32 at base+32..40 — same cacheline as the gathers
    float* f1 = (float*)&p1->f;
    float* f2 = (float*)&p2->f;
    atomicAdd(f1 + 0,  fx);
    atomicAdd(f1 + 1,  fy);
    atomicAdd(f1 + 2,  fz);
    atomicAdd(f2 + 0, -fx);
    atomicAdd(f2 + 1, -fy);
    atomicAdd(f2 + 2, -fz);
}

// ------------------------------------------------------------------
// Kernel 2: integrate vertices (coalesced b128s); zero f in place for
// the next substep. inv_mass comes from x.w — no mass load, no divide.
// ------------------------------------------------------------------
__global__ __launch_bounds__(256) void sm_integrate(Vert* __restrict__ W) {
    int i = blockIdx.x * blockDim.x + threadIdx.x;
    if (i >= NV) return;

    const float drag = expf(-DT * DRAG_DAMPING);   // constant-folded

    Vert* p = W + i;
    float4 x = p->x;
    float4 v = p->v;
    float4 f = p->f;
    float inv_m = x.w;

    // v = (v + DT * (F/m + g)) * drag   with g = (0,0,-9.8)
    float vx = (v.x + DT * (f.x * inv_m)) * drag;
    float vy = (v.y + DT * (f.y * inv_m)) * drag;
    float vz = (v.z + DT * (f.z * inv_m - 9.8f)) * drag;

    float nx = x.x + DT * vx;
    float ny = x.y + DT * vy;
    float nz = x.z + DT * vz;

    // ground plane: x.z = max(x.z, 0); v.z = 0 where x.z == 0
    nz = fmaxf(nz, 0.0f);
    if (nz == 0.0f) vz = 0.0f;

    p->x = make_float4(nx, ny, nz, inv_m);         // keep inv_mass in w
    p->v = make_float4(vx, vy, vz, 0.0f);
    p->f = make_float4(0.f, 0.f, 0.f, 0.f);        // reset accumulator
}

// ------------------------------------------------------------------
// Kernel 3: pack AoS positions into tight (NV,3) f32 output.
// ------------------------------------------------------------------
__global__ __launch_bounds__(256) void sm_pack(const Vert* __restrict__ W,
                                               float* __restrict__ out) {
    int i = blockIdx.x * blockDim.x + threadIdx.x;
    if (i >= NV) return;
    float4 p = W[i].x;
    out[3 * i + 0] = p.x;
    out[3 * i + 1] = p.y;
    out[3 * i + 2] = p.z;
}

extern "C" void kernel_launch(void* const* d_in, const int* in_sizes, int n_in,
                              void* d_out, int out_size, void* d_ws, size_t ws_size,
                              hipStream_t stream) {
    const float* x0      = (const float*)d_in[0];  // (NV,3) f32
    const int*   springs = (const int*)d_in[1];    // (NS,2) int32
    const float* rest    = (const float*)d_in[2];  // (NS,)  f32
    const float* mass    = (const float*)d_in[3];  // (NV,)  f32

    Vert* W = (Vert*)d_ws;                         // NV * 64B = 6.4 MB

    const int vert_blocks   = (NV + 255) / 256;
    const int spring_blocks = (NS + 255) / 256;

    sm_init<<<vert_blocks, 256, 0, stream>>>(x0, mass, W);

    for (int step = 0; step < NUM_SUBSTEPS; ++step) {
        sm_springs<<<spring_blocks, 256, 0, stream>>>(W, springs, rest);
        sm_integrate<<<vert_blocks, 256, 0, stream>>>(W);
    }

    sm_pack<<<vert_blocks, 256, 0, stream>>>(W, (float*)d_out);
}